// HybridQFCModel_65481071402331
// MI455X (gfx1250) — compile-verified
//
#include <hip/hip_runtime.h>
#include <math.h>

// ---------------------------------------------------------------------------
// HybridQFCModel on MI455X (gfx1250, wave32)
//   pool(6x6)->4 angles -> <Z_i> = r^T A_i r = diag(R A_i R^T)
//   -> BatchNorm (training stats) over batch of 131072.
// Memory-bound: TDM (TENSOR_LOAD_TO_LDS) DMAs only rows 0..5 x cols 0..23 of
// each 28x28 image into LDS (one descriptor per wave, 16 images each), then
// two chained V_WMMA_F32_16X16X4_F32 GEMMs per channel (32 WMMA / wave).
// ---------------------------------------------------------------------------

#define N_IMG   131072
#define IMG_PIX 784
#define NBLK2   2048            // N_IMG / (4 waves * 16 imgs)

typedef float v2f  __attribute__((ext_vector_type(2)));
typedef float v8f  __attribute__((ext_vector_type(8)));
typedef unsigned int u32x4 __attribute__((ext_vector_type(4)));
typedef int   i32x8 __attribute__((ext_vector_type(8)));
typedef int   i32x4 __attribute__((ext_vector_type(4)));

// ====================== kernel 1: build A_i matrices =======================
// One wave. Lane j owns column j of U (16 complex amps in registers).
// U = (RY)(CNOT chain)(RZ) x2 applied to identity columns.
// V[:,j] = U[:,j] * (-i)^popcount(j);  A_i[j][k] = sum_m sign_i(m) *
//   Re(V[m][j] * conj(V[m][k]))   (real symmetric; r^T A_i r == <Z_i>).
__global__ void qc_precompute(const float* __restrict__ qp,
                              float* __restrict__ Aw) {
  const int j = threadIdx.x;
  float ar[16], ai[16];
#pragma unroll
  for (int m = 0; m < 16; ++m) { ar[m] = (m == (j & 15)) ? 1.f : 0.f; ai[m] = 0.f; }

#pragma unroll
  for (int l = 0; l < 2; ++l) {
    // ---- RZ layer: diagonal phase exp(i * sum_w s_w * a_w), s_w = +-1 ----
    const float a0 = 0.5f * qp[(l * 4 + 0) * 2 + 0];
    const float a1 = 0.5f * qp[(l * 4 + 1) * 2 + 0];
    const float a2 = 0.5f * qp[(l * 4 + 2) * 2 + 0];
    const float a3 = 0.5f * qp[(l * 4 + 3) * 2 + 0];
#pragma unroll
    for (int m = 0; m < 16; ++m) {
      const float T = ((m & 8) ? a0 : -a0) + ((m & 4) ? a1 : -a1) +
                      ((m & 2) ? a2 : -a2) + ((m & 1) ? a3 : -a3);
      float pr, pi;  __sincosf(T, &pi, &pr);   // pr=cos, pi=sin
      const float nr = ar[m] * pr - ai[m] * pi;
      const float ni = ar[m] * pi + ai[m] * pr;
      ar[m] = nr; ai[m] = ni;
    }
    // ---- CNOT chain (c=0..2, t=c+1): swap rows (m, m|tm) where control set
#pragma unroll
    for (int c = 0; c < 3; ++c) {
      const int cm = 8 >> c, tm = 4 >> c;
#pragma unroll
      for (int m = 0; m < 16; ++m) {
        if ((m & cm) && !(m & tm)) {
          const int m1 = m | tm;
          const float tr = ar[m], ti = ai[m];
          ar[m] = ar[m1]; ai[m] = ai[m1];
          ar[m1] = tr;    ai[m1] = ti;
        }
      }
    }
    // ---- RY layer: [[c,-s],[s,c]] on each wire ----
#pragma unroll
    for (int w = 0; w < 4; ++w) {
      const float b = 0.5f * qp[(l * 4 + w) * 2 + 1];
      float sb, cb;  __sincosf(b, &sb, &cb);
      const int wm = 8 >> w;
#pragma unroll
      for (int m = 0; m < 16; ++m) {
        if (!(m & wm)) {
          const int m1 = m | wm;
          const float t0r = ar[m], t0i = ai[m], t1r = ar[m1], t1i = ai[m1];
          ar[m]  = cb * t0r - sb * t1r;  ai[m]  = cb * t0i - sb * t1i;
          ar[m1] = sb * t0r + cb * t1r;  ai[m1] = sb * t0i + cb * t1i;
        }
      }
    }
  }

  // fold initial-state phase D_j = (-i)^popcount(j) into this column
  const int pc = __popc(j & 15) & 3;
  const float djr = (pc == 0 ? 1.f : 0.f) - (pc == 2 ? 1.f : 0.f);
  const float dji = (pc == 3 ? 1.f : 0.f) - (pc == 1 ? 1.f : 0.f);
#pragma unroll
  for (int m = 0; m < 16; ++m) {
    const float nr = ar[m] * djr - ai[m] * dji;
    const float ni = ar[m] * dji + ai[m] * djr;
    ar[m] = nr; ai[m] = ni;
  }

  // A_i[j][k] via lane shuffles (lane k holds V[:,k])
  for (int k = 0; k < 16; ++k) {
    float acc0 = 0.f, acc1 = 0.f, acc2 = 0.f, acc3 = 0.f;
#pragma unroll
    for (int m = 0; m < 16; ++m) {
      const float wr = __shfl(ar[m], k);
      const float wi = __shfl(ai[m], k);
      const float p = ar[m] * wr + ai[m] * wi;   // Re(V[m][j]*conj(V[m][k]))
      acc0 += (m & 8) ? -p : p;
      acc1 += (m & 4) ? -p : p;
      acc2 += (m & 2) ? -p : p;
      acc3 += (m & 1) ? -p : p;
    }
    if (j < 16) {
      Aw[(0 * 16 + j) * 16 + k] = acc0;
      Aw[(1 * 16 + j) * 16 + k] = acc1;
      Aw[(2 * 16 + j) * 16 + k] = acc2;
      Aw[(3 * 16 + j) * 16 + k] = acc3;
    }
  }
}

// ====================== kernel 2: TDM + pool + double-WMMA + sums ==========
// Block = 128 thr (4 waves); each wave -> 16 images via one TDM descriptor:
//   3D tile [z=16 imgs][y=6 rows][x=24 cols], dim0 stride 28, dim1 stride 784.
// WMMA layouts (ISA 7.12.2, 32-bit):
//   A 16x4 : V0 = K0 | K2  (lanes 0-15 | 16-31),  V1 = K1 | K3, M = lane%16
//   B 4x16 : assumed mirror of A (transpose): V0 = rows K0|K2, V1 = K1|K3
//   D 16x16: VGPR v = row v (lanes 0-15) | row v+8 (lanes 16-31), N = lane%16
__global__ void __launch_bounds__(128) qpool_wmma(
    const float* __restrict__ x, const float* __restrict__ Aw,
    float* __restrict__ zout, float* __restrict__ blockSums) {
  __shared__ float stage[4][16][6][24];   // 36 KB: TDM destination
  __shared__ float r_lds[4][16][16];      //  4 KB
  __shared__ float wsum[4][8];

  const int tid   = threadIdx.x;
  const int wv    = tid >> 5;
  const int l     = tid & 31;
  const int half2 = l >> 4;            // which 16-lane half
  const int lm    = l & 15;
  const int img_base = (blockIdx.x * 4 + wv) * 16;

  // ---- issue TDM: stage 16 images x 6 rows x 24 cols into this wave's LDS
  {
    const int wvu = __builtin_amdgcn_readfirstlane(wv);
    const unsigned ibu = (unsigned)__builtin_amdgcn_readfirstlane(img_base);
    const unsigned long long ga =
        (unsigned long long)(uintptr_t)x + (unsigned long long)ibu * (IMG_PIX * 4u);
    const unsigned lds_base = (unsigned)(uintptr_t)&stage[wvu][0][0][0];

    u32x4 g0;
    g0.x = 1u;                                  // count=1 valid descriptor
    g0.y = lds_base;                            // lds_addr (bytes)
    g0.z = (unsigned)ga;                        // global_addr[31:0]
    g0.w = (unsigned)((ga >> 32) & 0x1FFFFFFu)  // global_addr[56:32]
           | (2u << 30);                        // type=2 ("image")
    i32x8 g1;
    g1[0] = 2 << 16;                            // data_size=4B
    g1[1] = 28 << 16;                           // tensor_dim0 = 28 (cols)
    g1[2] = 28 << 16;                           // tensor_dim1 = 28 (rows)
    g1[3] = 24 << 16;                           // tile_dim0 = 24
    g1[4] = 6 | (16 << 16);                     // tile_dim1=6, tile_dim2=16
    g1[5] = 28;                                 // tensor_dim0_stride = 28
    g1[6] = 784 << 16;                          // tensor_dim1_stride = 784
    g1[7] = 0;
    i32x4 g2; g2[0] = N_IMG; g2[1] = 0; g2[2] = 0; g2[3] = 0;   // tensor_dim2
    i32x4 g3; g3[0] = 0; g3[1] = 0; g3[2] = 0; g3[3] = 0;
    i32x8 g4; g4[0] = 0; g4[1] = 0; g4[2] = 0; g4[3] = 0;
              g4[4] = 0; g4[5] = 0; g4[6] = 0; g4[7] = 0;
    // clang-23 / therock arity: (v4u g0, v8i g1, v4i g2, v4i g3, v8i, cpol)
    __builtin_amdgcn_tensor_load_to_lds(g0, g1, g2, g3, g4, 0);
  }

  // ---- batch-invariant A-operand (overlaps TDM): rows of A_i, WMMA-A layout
  // AA[i][t] = (A_i[lm][4t+2*half2], A_i[lm][4t+2*half2+1])
  v2f AA[4][4];
#pragma unroll
  for (int i = 0; i < 4; ++i)
#pragma unroll
    for (int t = 0; t < 4; ++t)
      AA[i][t] = *(const v2f*)&Aw[i * 256 + lm * 16 + 4 * t + 2 * half2];

  __builtin_amdgcn_s_wait_tensorcnt(0);
  asm volatile("" ::: "memory");

  // ---- 6x6 avg-pool from staged rows (lane: image lm, rows 3*half2..+2) --
  float p0 = 0.f, p1 = 0.f, p2 = 0.f, p3 = 0.f;
  const float* srow = &stage[wv][lm][half2 * 3][0];
#pragma unroll
  for (int r = 0; r < 3; ++r) {
    const float4* rp = (const float4*)(srow + r * 24);
    const float4 f0 = rp[0], f1 = rp[1], f2 = rp[2];
    const float4 f3 = rp[3], f4v = rp[4], f5 = rp[5];
    p0 += f0.x + f0.y + f0.z + f0.w + f1.x + f1.y;
    p1 += f1.z + f1.w + f2.x + f2.y + f2.z + f2.w;
    p2 += f3.x + f3.y + f3.z + f3.w + f4v.x + f4v.y;
    p3 += f4v.z + f4v.w + f5.x + f5.y + f5.z + f5.w;
  }
  p0 += __shfl_xor(p0, 16);  p1 += __shfl_xor(p1, 16);
  p2 += __shfl_xor(p2, 16);  p3 += __shfl_xor(p3, 16);

  // angles |theta/2| < ~0.5 -> hardware v_sin/v_cos are accurate here
  const float sc = 0.5f / 36.f;
  const float a0 = p0 * sc, a1 = p1 * sc, a2 = p2 * sc, a3 = p3 * sc;
  const float c0 = __cosf(a0), s0 = __sinf(a0);
  const float c1 = __cosf(a1), s1 = __sinf(a1);
  const float c2 = __cosf(a2), s2 = __sinf(a2);
  const float c3 = __cosf(a3), s3 = __sinf(a3);

  if (l < 16) {
#pragma unroll
    for (int m = 0; m < 16; ++m) {
      const float rv = ((m & 8) ? s0 : c0) * ((m & 4) ? s1 : c1) *
                       ((m & 2) ? s2 : c2) * ((m & 1) ? s3 : c3);
      r_lds[wv][lm][m] = rv;
    }
  }
  __syncthreads();

  // ---- R tile registers: A-op of GEMM2 and B-op (=R^T) of GEMM1 ----------
  v2f Av[4];
#pragma unroll
  for (int t = 0; t < 4; ++t)
    Av[t] = *(const v2f*)&r_lds[wv][lm][4 * t + 2 * half2];

  float z[4] = {0.f, 0.f, 0.f, 0.f};
#pragma unroll
  for (int i = 0; i < 4; ++i) {
    // GEMM1: G_i = A_i * R^T   (D layout: lane l holds G_i[v+8*half2][lm])
    v8f g = {0.f, 0.f, 0.f, 0.f, 0.f, 0.f, 0.f, 0.f};
#pragma unroll
    for (int t = 0; t < 4; ++t)
      g = __builtin_amdgcn_wmma_f32_16x16x4_f32(
              false, AA[i][t], false, Av[t], (short)0, g, false, false);

    // half-swap so each lane can assemble G_i rows needed for the B-op
    float Gx[8];
#pragma unroll
    for (int v = 0; v < 8; ++v) Gx[v] = __shfl_xor(g[v], 16);

    // GEMM2: Z_i = R * G_i ; B-op slot t needs rows j0=4t+2*half2, j0+1.
    // row j lives in-lane if (j>=8)==half2 (reg j&7), else in Gx.
    v8f zt = {0.f, 0.f, 0.f, 0.f, 0.f, 0.f, 0.f, 0.f};
#pragma unroll
    for (int t = 0; t < 4; ++t) {
      const float b0_h0 = (4 * t     < 8) ?  g[4 * t    ] : Gx[4 * t - 8];
      const float b1_h0 = (4 * t + 1 < 8) ?  g[4 * t + 1] : Gx[4 * t - 7];
      const float b0_h1 = (4 * t + 2 < 8) ? Gx[4 * t + 2] :  g[4 * t - 6];
      const float b1_h1 = (4 * t + 3 < 8) ? Gx[4 * t + 3] :  g[4 * t - 5];
      v2f bb;
      bb.x = half2 ? b0_h1 : b0_h0;
      bb.y = half2 ? b1_h1 : b1_h0;
      zt = __builtin_amdgcn_wmma_f32_16x16x4_f32(
               false, Av[t], false, bb, (short)0, zt, false, false);
    }
    // diagonal: Z_i[b][b]; lane with lm == v+8*half2 holds it in zt[v]
#pragma unroll
    for (int v = 0; v < 8; ++v)
      if (lm == v + 8 * half2) z[i] = zt[v];
  }

  // holder lanes: (lm>>3)==half2  ->  lanes 0-7 (b=0..7), 24-31 (b=8..15)
  const bool holder = ((lm >> 3) == half2);
  if (holder) {
    float4 o;  o.x = z[0]; o.y = z[1]; o.z = z[2]; o.w = z[3];
    ((float4*)zout)[img_base + lm] = o;          // b = lm
  }

  // deterministic per-wave (sum, sumsq) tree-reduce in fp32 (16 terms each)
  float ls[8];
#pragma unroll
  for (int c = 0; c < 4; ++c) { ls[c] = z[c]; ls[4 + c] = z[c] * z[c]; }
#pragma unroll
  for (int c = 0; c < 8; ++c)
#pragma unroll
    for (int st = 1; st < 32; st <<= 1) ls[c] += __shfl_xor(ls[c], st);
  if (l == 0) {
#pragma unroll
    for (int c = 0; c < 8; ++c) wsum[wv][c] = ls[c];
  }
  __syncthreads();
  if (tid < 8) {
    float s = 0.f;
#pragma unroll
    for (int w = 0; w < 4; ++w) s += wsum[w][tid];
    blockSums[blockIdx.x * 8 + tid] = s;
  }
}

// ====================== kernel 3: finalize BN scale/bias ===================
__global__ void bn_stats(const float* __restrict__ blockSums,
                         const float* __restrict__ gamma,
                         const float* __restrict__ beta,
                         float* __restrict__ stats) {
  __shared__ double tot[8];
  const int tid = threadIdx.x;
  const int wv = tid >> 5, l = tid & 31;
  double s = 0;
  for (int t = l; t < NBLK2; t += 32) s += (double)blockSums[t * 8 + wv];
#pragma unroll
  for (int st = 1; st < 32; st <<= 1) s += __shfl_xor(s, st);
  if (l == 0) tot[wv] = s;
  __syncthreads();
  if (tid < 4) {
    const double n = (double)N_IMG;
    const double mean = tot[tid] / n;
    const double var  = tot[4 + tid] / n - mean * mean;   // biased, in double
    const double inv  = 1.0 / sqrt(var + 1e-5);
    const float scale = (float)inv * gamma[tid];
    const float bias  = beta[tid] - (float)mean * scale;
    stats[tid]     = scale;
    stats[4 + tid] = bias;
  }
}

// ====================== kernel 4: apply BN in place ========================
__global__ void __launch_bounds__(256) bn_apply(float* __restrict__ out,
                                                const float* __restrict__ stats) {
  const int idx = blockIdx.x * 256 + threadIdx.x;
  const int c = idx & 3;
  out[idx] = out[idx] * stats[c] + stats[4 + c];
}

// ===========================================================================
extern "C" void kernel_launch(void* const* d_in, const int* in_sizes, int n_in,
                              void* d_out, int out_size, void* d_ws, size_t ws_size,
                              hipStream_t stream) {
  (void)in_sizes; (void)n_in; (void)out_size; (void)ws_size;
  const float* x     = (const float*)d_in[0];
  const float* qp    = (const float*)d_in[1];
  const float* gamma = (const float*)d_in[2];
  const float* beta  = (const float*)d_in[3];

  float* Aw        = (float*)d_ws;                                   // 4 KB
  float* blockSums = (float*)((char*)d_ws + 4096);                   // 64 KB
  float* stats     = (float*)((char*)d_ws + 4096 + NBLK2 * 8 * sizeof(float));
  float* zout      = (float*)d_out;

  hipLaunchKernelGGL(qc_precompute, dim3(1), dim3(32), 0, stream, qp, Aw);
  hipLaunchKernelGGL(qpool_wmma, dim3(NBLK2), dim3(128), 0, stream,
                     x, Aw, zout, blockSums);
  hipLaunchKernelGGL(bn_stats, dim3(1), dim3(256), 0, stream,
                     blockSums, gamma, beta, stats);
  hipLaunchKernelGGL(bn_apply, dim3((N_IMG * 4) / 256), dim3(256), 0, stream,
                     zout, stats);
}